// MaskAlgorythmRaw_7919919694474
// MI455X (gfx1250) — compile-verified
//
#include <hip/hip_runtime.h>
#include <hip/hip_bf16.h>

typedef unsigned int uint;
typedef unsigned long long u64;
typedef __attribute__((ext_vector_type(4))) float v4f;  // native 128-bit vector

// ---------------- problem constants (from reference) ----------------
#define Bn 1024
#define Pn 512
#define Kn 128
#define MAXC 76          // int(0.15*4096/8)
#define NUM_MASK 819     // int(1024*0.8)
#define NUM_RAND 102     // int(1024*0.1)
#define TILE_P 64        // patches staged per block in kernel C

// ---------------- JAX threefry2x32 (20 rounds) ----------------
__device__ __forceinline__ uint rotl32(uint x, uint r) {
  return (x << r) | (x >> (32u - r));
}

__device__ __forceinline__ void threefry2x32(uint k0, uint k1, uint c0, uint c1,
                                             uint& o0, uint& o1) {
  uint ks2 = k0 ^ k1 ^ 0x1BD11BDAu;
  uint x0 = c0 + k0;
  uint x1 = c1 + k1;
#define TF_R(r) { x0 += x1; x1 = rotl32(x1, r); x1 ^= x0; }
  TF_R(13) TF_R(15) TF_R(26) TF_R(6)
  x0 += k1;  x1 += ks2 + 1u;
  TF_R(17) TF_R(29) TF_R(16) TF_R(24)
  x0 += ks2; x1 += k0 + 2u;
  TF_R(13) TF_R(15) TF_R(26) TF_R(6)
  x0 += k0;  x1 += k1 + 3u;
  TF_R(17) TF_R(29) TF_R(16) TF_R(24)
  x0 += k1;  x1 += ks2 + 4u;
  TF_R(13) TF_R(15) TF_R(26) TF_R(6)
  x0 += ks2; x1 += k0 + 5u;
#undef TF_R
  o0 = x0; o1 = x1;
}

// JAX random_bits pairing: n elements, counter halves (i, i+half)
__device__ __forceinline__ uint jax_randbits(uint k0, uint k1, uint i, uint half) {
  uint c0, c1, o0, o1;
  if (i < half) { c0 = i; c1 = i + half; } else { c0 = i - half; c1 = i; }
  threefry2x32(k0, k1, c0, c1, o0, o1);
  return (i < half) ? o0 : o1;
}

// JAX uniform [0,1): (bits>>9)|0x3f800000 bitcast - 1
__device__ __forceinline__ float jax_u01(uint bits) {
  return __uint_as_float((bits >> 9) | 0x3f800000u) - 1.0f;
}

// ---------------- CDNA5 async global -> LDS (ASYNCcnt path) ----------------
__device__ __forceinline__ void async_load_b128(uint lds_byte_off, const void* gaddr) {
  asm volatile("global_load_async_to_lds_b128 %0, %1, off"
               :
               : "v"(lds_byte_off), "v"((u64)(size_t)gaddr)
               : "memory");
}
__device__ __forceinline__ void wait_asynccnt0() {
  asm volatile("s_wait_asynccnt 0x0" ::: "memory");
}

// =====================================================================
// Kernel A: batch permutation -> mask_type (1 block, 1024 threads)
// Derives kp/ks/kr from root key (0,42); rank-based stable argsort of
// random bits gives each batch index its permutation rank -> group.
// =====================================================================
__global__ __launch_bounds__(1024) void mask_type_kernel(float* __restrict__ mt_out,
                                                         uint* __restrict__ hdr) {
  __shared__ u64 keys[Bn];
  const int j = threadIdx.x;

  // split(key(42), 3): threefry(key, iota(6)) pairs (0,3),(1,4),(2,5)
  uint o0a, o1a, o0b, o1b, o0c, o1c;
  threefry2x32(0u, 42u, 0u, 3u, o0a, o1a);
  threefry2x32(0u, 42u, 1u, 4u, o0b, o1b);
  threefry2x32(0u, 42u, 2u, 5u, o0c, o1c);
  const uint kp0 = o0a, kp1 = o0b;      // kp = (out[0], out[1])
  const uint ks0 = o0c, ks1 = o1a;      // ks = (out[2], out[3])
  const uint kr0 = o1b, kr1 = o1c;      // kr = (out[4], out[5])

  const uint bits = jax_randbits(kp0, kp1, (uint)j, Bn / 2u);
  const u64 mine = ((u64)bits << 32) | (uint)j;  // idx tiebreak = stable sort
  keys[j] = mine;
  __syncthreads();

  int rank = 0;
  for (int q = 0; q < Bn; ++q) rank += (keys[q] < mine) ? 1 : 0;

  const uint g = (rank < NUM_MASK) ? 0u : ((rank < NUM_MASK + NUM_RAND) ? 1u : 2u);
  mt_out[j] = (float)g;
  if (j == 0) {
    hdr[0] = g;            // t0 = mask_type[0]
    hdr[1] = ks0; hdr[2] = ks1;
    hdr[3] = kr0; hdr[4] = kr1;
  }
}

// =====================================================================
// Kernel B: per-row corruption selection (1024 blocks x 512 threads)
// scores = uniform(ks); invalid patches = +inf; stable argsort via
// (bits,idx)-packed rank; sel = rank < n_corr; emits I row and sel bitmask.
// =====================================================================
__global__ __launch_bounds__(512) void select_kernel(const int* __restrict__ seq_len,
                                                     float* __restrict__ I_out,
                                                     const uint* __restrict__ hdr,
                                                     uint* __restrict__ sel_bits) {
  __shared__ u64 keys[Pn];
  __shared__ uint words[Pn / 32];
  const int b = blockIdx.x;
  const int p = threadIdx.x;

  const uint ks0 = hdr[1], ks1 = hdr[2];
  const int s = seq_len[b];
  const int n_valid = s >> 3;                                   // seq_len // 8
  const int n_corr = (int)floorf(0.15f * (float)s / 8.0f);      // floor(0.15*s/8)

  const uint i = (uint)(b * Pn + p);
  const uint bits = jax_randbits(ks0, ks1, i, (uint)((Bn * Pn) / 2));
  const float u = jax_u01(bits);
  const uint fb = (p < n_valid) ? __float_as_uint(u) : 0x7F800000u;  // +inf pad
  const u64 key = ((u64)fb << 32) | (uint)p;
  keys[p] = key;
  if (p < Pn / 32) words[p] = 0u;
  __syncthreads();

  int rank = 0;
  for (int q = 0; q < Pn; ++q) rank += (keys[q] < key) ? 1 : 0;
  const bool sel = rank < n_corr;

  if (sel) atomicOr(&words[p >> 5], 1u << (p & 31));

  float* Irow = I_out + (size_t)b * MAXC;
  if (p < MAXC) Irow[p] = -1.0f;
  __syncthreads();

  if (p < Pn / 32) sel_bits[b * (Pn / 32) + p] = words[p];
  if (sel && rank < MAXC) Irow[rank] = (float)p;  // I[b][rank] = order slot
}

// =====================================================================
// Kernel C: streaming apply (HBM-bound, ~512MB @ 23.3TB/s ~= 22us floor)
// grid (8,128), 256 thr. Each block: async-stage 32KB pos tile + w_mask
// into LDS once (global_load_async_to_lds_b128 / s_wait_asynccnt), then
// loop 8 batch rows. One wave32 == one patch row (128 f32 = 32 lanes x
// b128) -> sel branch is wave-uniform. x loads and out stores use
// non-temporal hints so the 512MB stream does not thrash the 192MB L2.
// =====================================================================
__global__ __launch_bounds__(256) void mask_apply_kernel(const float* __restrict__ x,
                                                         const float* __restrict__ pos,
                                                         const float* __restrict__ wm,
                                                         float* __restrict__ out,
                                                         const uint* __restrict__ hdr,
                                                         const uint* __restrict__ sel_bits) {
  __shared__ v4f sp[TILE_P * (Kn / 4)];     // 32 KB pos tile
  __shared__ v4f swm[Kn / 4];               // 512 B w_mask
  __shared__ uint s_hdr[5];

  const int tid = threadIdx.x;
  const int p0 = blockIdx.x * TILE_P;

  // ---- async stage pos[p0:p0+64, :] (contiguous 32KB) + w_mask into LDS ----
  {
    const float* src = pos + (size_t)p0 * Kn;
    const uint lds_base = (uint)(size_t)(&sp[0]);
    for (int v = tid; v < TILE_P * (Kn / 4); v += 256)
      async_load_b128(lds_base + (uint)v * 16u, src + (size_t)v * 4);
    if (tid < Kn / 4) {
      const uint lb = (uint)(size_t)(&swm[0]);
      async_load_b128(lb + (uint)tid * 16u, wm + (size_t)tid * 4);
    }
    if (tid < 5) s_hdr[tid] = hdr[tid];
    wait_asynccnt0();       // drain this wave's ASYNCcnt
    __syncthreads();        // make all waves' LDS writes visible
  }
  const uint t0 = s_hdr[0];
  const uint kr0 = s_hdr[3], kr1 = s_hdr[4];

  for (int it = 0; it < Bn / 128; ++it) {
    const int b = (int)blockIdx.y * (Bn / 128) + it;
    const v4f* __restrict__ xb = (const v4f*)x + ((size_t)b * Pn + p0) * (Kn / 4);
    v4f* __restrict__ ob = (v4f*)out + ((size_t)b * Pn + p0) * (Kn / 4);
    const uint* __restrict__ selw = sel_bits + (size_t)b * (Pn / 32) + (p0 >> 5);

    // hoist this row's two selection-bitmask words into registers
    const uint sw0 = selw[0];
    const uint sw1 = selw[1];

    if (it + 1 < Bn / 128)  // deepen HBM pipeline: prefetch next row's tile
      __builtin_prefetch((const void*)(xb + (size_t)Pn * (Kn / 4)), 0, 0);

    for (int v = tid; v < TILE_P * (Kn / 4); v += 256) {
      const int pl = v >> 5;   // patch within tile (uniform across a wave32)
      const int k4 = v & 31;   // float4 index within K
      v4f xv = __builtin_nontemporal_load(xb + v);     // th:NT streaming read
      const uint sword = (pl < 32) ? sw0 : sw1;
      const bool sel = (sword >> (pl & 31)) & 1u;
      v4f ov = xv;
      if (t0 == 0u) {          // repl = w_mask + pos
        if (sel) {
          const v4f pv = sp[v];
          const v4f wv = swm[k4];
          ov = wv + pv;
        }
      } else if (t0 == 1u) {   // repl = uniform(kr) + pos  (threefry only if sel)
        if (sel) {
          const v4f pv = sp[v];
          const uint base = ((uint)(b * Pn + (p0 + pl)) << 7) + ((uint)k4 << 2);
          const uint half = 1u << 25;  // n = B*P*K = 2^26
          ov.x = jax_u01(jax_randbits(kr0, kr1, base + 0u, half)) + pv.x;
          ov.y = jax_u01(jax_randbits(kr0, kr1, base + 1u, half)) + pv.y;
          ov.z = jax_u01(jax_randbits(kr0, kr1, base + 2u, half)) + pv.z;
          ov.w = jax_u01(jax_randbits(kr0, kr1, base + 3u, half)) + pv.w;
        }
      }                        // t0 == 2: repl = x -> plain copy
      __builtin_nontemporal_store(ov, ob + v);         // th:NT streaming write
    }
  }
}

// =====================================================================
extern "C" void kernel_launch(void* const* d_in, const int* in_sizes, int n_in,
                              void* d_out, int out_size, void* d_ws, size_t ws_size,
                              hipStream_t stream) {
  const float* x   = (const float*)d_in[0];
  const float* pos = (const float*)d_in[1];
  const float* wm  = (const float*)d_in[2];
  const int* seq   = (const int*)d_in[3];

  float* out = (float*)d_out;
  float* I_out  = out + (size_t)Bn * Pn * Kn;        // [B, 76] as float
  float* mt_out = I_out + (size_t)Bn * MAXC;         // [B]

  uint* ws  = (uint*)d_ws;
  uint* hdr = ws;                                    // [0]=t0 [1..2]=ks [3..4]=kr
  uint* sel = ws + 16;                               // B * 16 words of sel bits

  mask_type_kernel<<<dim3(1), dim3(1024), 0, stream>>>(mt_out, hdr);
  select_kernel<<<dim3(Bn), dim3(Pn), 0, stream>>>(seq, I_out, hdr, sel);
  mask_apply_kernel<<<dim3(Pn / TILE_P, 128), dim3(256), 0, stream>>>(
      x, pos, wm, out, hdr, sel);
}